// Spacial_IRNN_31628139168192
// MI455X (gfx1250) — compile-verified
//
#include <hip/hip_runtime.h>
#include <cstdint>

static constexpr int Bn = 4;
static constexpr int Cn = 64;
static constexpr int Hn = 256;
static constexpr int Wn = 256;
static constexpr int TP = 33;                 // padded LDS tile pitch (floats)
static constexpr size_t PLANE = (size_t)Bn * Cn * Hn * Wn;

// ---------------------------------------------------------------------------
// Vertical scans (up = dir 0, down = dir 2). Thread-per-column: threadIdx.x is
// the w coordinate -> every load/store step is a contiguous 1KB line per block.
// ---------------------------------------------------------------------------
__global__ __launch_bounds__(256) void irnn_vert(
    const float* __restrict__ x,
    const float* __restrict__ w_up,   const float* __restrict__ b_up,
    const float* __restrict__ w_down, const float* __restrict__ b_down,
    float* __restrict__ out)
{
    const int bc = blockIdx.x;            // 0 .. B*C-1
    const int c  = bc % Cn;
    const int w  = threadIdx.x;           // 0 .. W-1
    const size_t base = (size_t)bc * Hn * Wn + w;

    if (blockIdx.y == 0) {
        // down scan: h=0 boundary, increasing h. dir index 2
        const float wc = w_down[c];
        const float bcf = b_down[c];
        float* o = out + 2 * PLANE;
        size_t idx = base;
        float h = x[idx];
        o[idx] = h;                        // boundary = raw input
        #pragma unroll 4
        for (int hh = 1; hh < Hn; ++hh) {
            idx += Wn;
            const float xv = x[idx];
            h = fmaxf(fmaf(wc, h, bcf) + xv, 0.0f);
            o[idx] = h;
        }
    } else {
        // up scan: h=H-1 boundary, decreasing h. dir index 0
        const float wc = w_up[c];
        const float bcf = b_up[c];
        float* o = out;                    // plane 0
        size_t idx = base + (size_t)(Hn - 1) * Wn;
        float h = x[idx];
        o[idx] = h;
        #pragma unroll 4
        for (int hh = Hn - 2; hh >= 0; --hh) {
            idx -= Wn;
            const float xv = x[idx];
            h = fmaxf(fmaf(wc, h, bcf) + xv, 0.0f);
            o[idx] = h;
        }
    }
}

// ---------------------------------------------------------------------------
// Horizontal scans (right = dir 1, left = dir 3). One wave owns 32 consecutive
// rows; 32x32 fp32 tiles are staged through a private LDS slab using the CDNA5
// async global<->LDS data mover, scanned transposed in LDS, and drained back
// with async stores. No cross-wave synchronization needed.
// ---------------------------------------------------------------------------
__global__ __launch_bounds__(256) void irnn_horiz(
    const float* __restrict__ x,
    const float* __restrict__ w_right, const float* __restrict__ b_right,
    const float* __restrict__ w_left,  const float* __restrict__ b_left,
    float* __restrict__ out)
{
    __shared__ float tile_s[8][32 * TP];   // 8 waves/block, 4.2KB each

    const int lane = threadIdx.x & 31;     // wave32
    const int wave = threadIdx.x >> 5;
    const int rowBlock = blockIdx.x * 8 + wave;   // 0 .. 2047
    const int rbase = rowBlock * 32;              // global row = bc*H + h
    const int c = (rbase / Hn) % Cn;              // 32|H -> whole tile same (b,c)

    const bool rev = (blockIdx.y != 0);
    const float wc  = rev ? w_left[c] : w_right[c];
    const float bcf = rev ? b_left[c] : b_right[c];

    const float* gin = x + (size_t)rbase * Wn;
    float* gout = out + (rev ? (size_t)3 : (size_t)1) * PLANE + (size_t)rbase * Wn;

    float* tile = tile_s[wave];
    // LDS byte offset: flat LDS addresses carry the offset in addr[31:0]
    const unsigned lds_base = (unsigned)(uintptr_t)tile;

    float h = 0.0f;
    for (int cbk = 0; cbk < Wn / 32; ++cbk) {
        const int cbase = rev ? (Wn - 32 - cbk * 32) : (cbk * 32);

        // ---- async load 32x32 tile: instruction k moves row k (one 128B line)
        {
            unsigned long long g = (unsigned long long)(uintptr_t)(gin + cbase + lane);
            unsigned laddr = lds_base + (unsigned)(lane * 4);
            #pragma unroll
            for (int k = 0; k < 32; ++k) {
                asm volatile("global_load_async_to_lds_b32 %0, %1, off"
                             :: "v"(laddr), "v"(g) : "memory");
                g += (unsigned long long)Wn * 4;
                laddr += TP * 4;
            }
        }
        asm volatile("s_wait_asynccnt 0x0" ::: "memory");

        // ---- scan: lane scans its own LDS row (stride TP -> conflict-free)
        #pragma unroll
        for (int j = 0; j < 32; ++j) {
            const int jj = rev ? (31 - j) : j;
            const int gcol = cbase + jj;
            const float xv = tile[lane * TP + jj];
            const bool boundary = rev ? (gcol == Wn - 1) : (gcol == 0);
            h = boundary ? xv : fmaxf(fmaf(wc, h, bcf) + xv, 0.0f);
            tile[lane * TP + jj] = h;      // in-place: only this lane touches row
        }
        // make DS writes visible to the async engine before it reads LDS
        asm volatile("s_wait_dscnt 0x0" ::: "memory");

        // ---- async store tile back (coalesced 128B lines per instruction)
        {
            unsigned long long g = (unsigned long long)(uintptr_t)(gout + cbase + lane);
            unsigned laddr = lds_base + (unsigned)(lane * 4);
            #pragma unroll
            for (int k = 0; k < 32; ++k) {
                asm volatile("global_store_async_from_lds_b32 %0, %1, off"
                             :: "v"(g), "v"(laddr) : "memory");
                g += (unsigned long long)Wn * 4;
                laddr += TP * 4;
            }
        }
        // tile is reused next chunk: wait for stores to finish reading LDS
        asm volatile("s_wait_asynccnt 0x0" ::: "memory");
    }
}

// ---------------------------------------------------------------------------
extern "C" void kernel_launch(void* const* d_in, const int* in_sizes, int n_in,
                              void* d_out, int out_size, void* d_ws, size_t ws_size,
                              hipStream_t stream)
{
    const float* input   = (const float*)d_in[0];
    const float* w_up    = (const float*)d_in[1];
    const float* w_right = (const float*)d_in[2];
    const float* w_down  = (const float*)d_in[3];
    const float* w_left  = (const float*)d_in[4];
    const float* b_up    = (const float*)d_in[5];
    const float* b_right = (const float*)d_in[6];
    const float* b_down  = (const float*)d_in[7];
    const float* b_left  = (const float*)d_in[8];
    float* out = (float*)d_out;

    // planes 0 (up) and 2 (down)
    irnn_vert<<<dim3(Bn * Cn, 2), dim3(256), 0, stream>>>(
        input, w_up, b_up, w_down, b_down, out);

    // planes 1 (right) and 3 (left)
    irnn_horiz<<<dim3((Bn * Cn * Hn) / (32 * 8), 2), dim3(256), 0, stream>>>(
        input, w_right, b_right, w_left, b_left, out);
}